// HexagonalPatchExtractor_42760694399367
// MI455X (gfx1250) — compile-verified
//
#include <hip/hip_runtime.h>
#include <cstdint>

#define IMG   224
#define NCENT 340   // 20 cols * 17 rows of hex centers
#define NROWS 17
#define CH    3
#define GB    8     // images per block (B=128 -> gridDim.y = 16)

// Exact double value of numpy sqrt(3.0)*8.0 (round-trip literal).
#define ROW_SP 13.856406460551018

// Matches reference _axis_indices: s is exact in double (size<=16, /16 exact),
// so floor and the f32-cast weight match numpy bit-for-bit.
__device__ __forceinline__ void axis_tab(int p, int size, int& i0, int& i1, float& w) {
    double s = ((double)p + 0.5) * (double)size * 0.0625 - 0.5;
    if (s < 0.0) s = 0.0;
    double f = floor(s);
    int a  = (int)f;
    int bq = a + 1;
    int m  = size - 1;
    i0 = a;
    i1 = (bq < m) ? bq : m;
    w  = (float)(s - f);
}

__global__ __launch_bounds__(256)
void hexpatch_async_kernel(const float* __restrict__ images,
                           float* __restrict__ out, int B) {
    __shared__ float tile[2][CH * 256];  // double-buffered 3ch x 16x16 regions
    __shared__ int   s_i0[32];           // [0,16): x-axis, [16,32): y-axis
    __shared__ int   s_i1[32];
    __shared__ float s_w[32];

    const int n  = blockIdx.x;           // hex cell (no div/mod needed)
    const int b0 = blockIdx.y * GB;      // first image of this block's batch
    const int nb = min(GB, B - b0);
    const int t  = threadIdx.x;          // t = py*16 + px

    // --- geometry (uniform per block, amortized over nb images) ---
    const int col = n / NROWS;
    const int row = n - col * NROWS;
    double cx = (double)col * 12.0;
    double cy = (double)row * ROW_SP + ((col & 1) ? (ROW_SP * 0.5) : 0.0);
    int x1 = (int)fmax(0.0, cx - 8.0);
    int x2 = (int)fmin(224.0, cx + 8.0);
    int y1 = (int)fmax(0.0, cy - 8.0);
    int y2 = (int)fmin(224.0, cy + 8.0);
    int sx = x2 - x1;
    int sy = y2 - y1;

    const int ry  = t >> 4;
    const int rx  = t & 15;
    const int ryc = min(ry, sy - 1);     // clamp: every lane issues a valid addr
    const int rxc = min(rx, sx - 1);

    const size_t plane = (size_t)(IMG * IMG);
    const float* srcp = images + (size_t)b0 * (CH * plane)
                               + (size_t)(y1 + ryc) * IMG + (x1 + rxc);
    const uint32_t lds0     = (uint32_t)(uintptr_t)(&tile[0][0]);
    const uint32_t lane_lds = 4u * (uint32_t)t;

    auto stage = [&](int buf, int bi) {
        const float* sp = srcp + (size_t)bi * (CH * plane);
        uint32_t lb = lds0 + (uint32_t)(buf * (CH * 256 * 4)) + lane_lds;
#pragma unroll
        for (int c = 0; c < CH; ++c) {
            uint64_t ga = (uint64_t)(uintptr_t)(sp + (size_t)c * plane);
            uint32_t la = lb + (uint32_t)(c * 1024);
            asm volatile("global_load_async_to_lds_b32 %0, %1, off"
                         :: "v"(la), "v"(ga) : "memory");
        }
    };

    stage(0, 0);                         // prime the pipeline

    // --- bilinear tables: computed once per block by first 32 lanes ---
    if (t < 16) {
        int i0, i1; float w;
        axis_tab(t, sx, i0, i1, w);
        s_i0[t] = i0; s_i1[t] = i1; s_w[t] = w;
    } else if (t < 32) {
        int i0, i1; float w;
        axis_tab(t - 16, sy, i0, i1, w);
        s_i0[t] = i0; s_i1[t] = i1; s_w[t] = w;
    }

    float* obase = out + ((size_t)b0 * NCENT + n) * (CH * 256) + t;

    int i0x = 0, i1x = 0, i0y = 0, i1y = 0;
    float wx = 0.f, wy = 0.f, omwx = 0.f, omwy = 0.f;

    for (int i = 0; i < nb; ++i) {
        if (i + 1 < nb) {
            stage((i + 1) & 1, i + 1);                     // prefetch next image
            asm volatile("s_wait_asynccnt 0x3" ::: "memory"); // batch i landed
        } else {
            asm volatile("s_wait_asynccnt 0x0" ::: "memory");
        }
        __syncthreads();   // tile[i&1] visible to all waves; tables ready (i==0)

        if (i == 0) {
            i0x = s_i0[rx];      i1x = s_i1[rx];      wx = s_w[rx];
            i0y = s_i0[16 + ry]; i1y = s_i1[16 + ry]; wy = s_w[16 + ry];
            omwx = 1.0f - wx;
            omwy = 1.0f - wy;
        }

        const float* tl0 = &tile[i & 1][0];
        const int o00 = i0y * 16 + i0x;
        const int o01 = i0y * 16 + i1x;
        const int o10 = i1y * 16 + i0x;
        const int o11 = i1y * 16 + i1x;

        float* o = obase + (size_t)i * (NCENT * CH * 256);
#pragma unroll
        for (int c = 0; c < CH; ++c) {
            const float* tl = tl0 + c * 256;
            float v00 = tl[o00], v01 = tl[o01], v10 = tl[o10], v11 = tl[o11];
            float r = v00 * omwy * omwx + v01 * omwy * wx
                    + v10 * wy   * omwx + v11 * wy   * wx;
            __builtin_nontemporal_store(r, o + c * 256);  // write-once stream
        }

        if (i + 1 < nb) __syncthreads();  // all waves done reading tile[i&1]
                                          // before it is re-staged at i+2
    }
}

extern "C" void kernel_launch(void* const* d_in, const int* in_sizes, int n_in,
                              void* d_out, int out_size, void* d_ws, size_t ws_size,
                              hipStream_t stream) {
    (void)n_in; (void)out_size; (void)d_ws; (void)ws_size;
    const float* images = (const float*)d_in[0];
    float* out = (float*)d_out;
    int B = in_sizes[0] / (CH * IMG * IMG);   // 128 for the reference setup
    dim3 grid((unsigned)NCENT, (unsigned)((B + GB - 1) / GB));
    hexpatch_async_kernel<<<grid, 256, 0, stream>>>(images, out, B);
}